// GCPInteractions_37847251812666
// MI455X (gfx1250) — compile-verified
//
#include <hip/hip_runtime.h>
#include <cstddef>

typedef __attribute__((ext_vector_type(2))) float v2f;
typedef __attribute__((ext_vector_type(8))) float v8f;

constexpr int N_NODES = 16384;
constexpr int N_EDGES = 262144;
constexpr int C_S = 128;
constexpr int C_V = 16;
constexpr int E_SC = 32;
constexpr int E_VC = 4;

__device__ __forceinline__ float sigmoidf_(float x) { return 1.0f / (1.0f + __expf(-x)); }
__device__ __forceinline__ float siluf_(float x) { return x * sigmoidf_(x); }

// D = A(16x4) * B(4x16) + C, fp32 WMMA. Layout per CDNA5 ISA 7.12.2:
//  A: lane (hi,lo) vgpr j -> A[lo][2*hi+j];  B: -> B[2*hi+j][lo];  C/D vgpr i -> [i+8*hi][lo]
__device__ __forceinline__ v8f wmma4(float ax, float ay, float bx, float by, v8f c) {
  v2f a; a.x = ax; a.y = ay;
  v2f b; b.x = bx; b.y = by;
  return __builtin_amdgcn_wmma_f32_16x16x4_f32(false, a, false, b, (short)0, c, false, false);
}

// ---------------- LayerNorm (gcp_ln) ----------------
__global__ void k_layernorm(const float* __restrict__ s_in, const float* __restrict__ v_in,
                            const float* __restrict__ g, const float* __restrict__ b,
                            float* __restrict__ s_out, float* __restrict__ v_out) {
  __shared__ float red[C_S];
  const int n = blockIdx.x;
  const int t = threadIdx.x;
  float sv = s_in[(size_t)n * C_S + t];
  red[t] = sv; __syncthreads();
  for (int o = C_S / 2; o > 0; o >>= 1) { if (t < o) red[t] += red[t + o]; __syncthreads(); }
  float mu = red[0] / C_S; __syncthreads();
  float d0 = sv - mu;
  red[t] = d0 * d0; __syncthreads();
  for (int o = C_S / 2; o > 0; o >>= 1) { if (t < o) red[t] += red[t + o]; __syncthreads(); }
  float var = red[0] / C_S; __syncthreads();
  s_out[(size_t)n * C_S + t] = d0 * rsqrtf(var + 1e-5f) * g[t] + b[t];
  float vv = (t < C_V * 3) ? v_in[(size_t)n * C_V * 3 + t] : 0.0f;
  red[t] = vv * vv; __syncthreads();
  for (int o = C_S / 2; o > 0; o >>= 1) { if (t < o) red[t] += red[t + o]; __syncthreads(); }
  float vn2 = red[0] / C_V;
  if (t < C_V * 3) v_out[(size_t)n * C_V * 3 + t] = vv * rsqrtf(vn2 + 1e-5f);
}

__global__ void k_zero(float* __restrict__ p, long n) {
  long i = (long)blockIdx.x * blockDim.x + threadIdx.x;
  if (i < n) p[i] = 0.0f;
}

__global__ void k_counts(const int* __restrict__ ei, float* __restrict__ cnt_row,
                         float* __restrict__ cnt_col) {
  int e = blockIdx.x * blockDim.x + threadIdx.x;
  if (e >= N_EDGES) return;
  atomicAdd(&cnt_row[ei[e]], 1.0f);
  atomicAdd(&cnt_col[ei[N_EDGES + e]], 1.0f);
}

// ---------------- Message GCP 0 (fused, 32 edges / block, 2 M-tiles / wave) ----------------
constexpr int H0 = 36;                       // hidden vector channels
constexpr int K0 = 2 * C_S + E_SC + 4 * H0;  // 432
constexpr int SA0 = K0 + 4;                  // 436 = 4*odd: conflict-free, rows 16B aligned
constexpr int EB0 = 32;
constexpr int MT0 = EB0 / 16;

__global__ void __launch_bounds__(256)
k_msg0(const float* __restrict__ ln_s, const float* __restrict__ ln_v,
       const float* __restrict__ edge_s, const float* __restrict__ edge_v,
       const float* __restrict__ frames, const int* __restrict__ ei,
       const float* __restrict__ Wvd, const float* __restrict__ Ws,
       const float* __restrict__ bs, const float* __restrict__ Wvu,
       const float* __restrict__ Wg, const float* __restrict__ bg,
       float* __restrict__ rs, float* __restrict__ rv) {
  __shared__ __align__(16) float sA[EB0 * SA0];
  __shared__ float sWvd[H0 * H0];
  __shared__ float sMVNS[EB0 * 129];  // union: sMV (phase 1) / sNS (phase 2), disjoint live ranges
  __shared__ float sVH[EB0 * H0 * 3];
  __shared__ float sGate[EB0 * 16];
  __shared__ int sRow[EB0], sCol[EB0];
  float* sMV = sMVNS;
  float* sNS = sMVNS;
  const int t = threadIdx.x;
  const int e0 = blockIdx.x * EB0;
  if (t < EB0) { sRow[t] = ei[e0 + t]; sCol[t] = ei[N_EDGES + e0 + t]; }
  for (int i = t; i < H0 * H0; i += 256) sWvd[i] = Wvd[i];
  __syncthreads();
  // ms = [ln_s[row] | edge_s | ln_s[col]] -> sA[:, 0:288], float4 vectorized (72 quads/row)
  for (int i = t; i < EB0 * 72; i += 256) {
    int e = i / 72, q = i % 72;
    float4 val;
    if (q < 32)      val = *(const float4*)(ln_s + (size_t)sRow[e] * C_S + 4 * q);
    else if (q < 40) val = *(const float4*)(edge_s + (size_t)(e0 + e) * E_SC + 4 * (q - 32));
    else             val = *(const float4*)(ln_s + (size_t)sCol[e] * C_S + 4 * (q - 40));
    *(float4*)(&sA[e * SA0 + 4 * q]) = val;
  }
  // mv = [ln_v[row] | edge_v | ln_v[col]]  (36 channels x 3)
  for (int i = t; i < EB0 * H0 * 3; i += 256) {
    int e = i / (H0 * 3), r = i % (H0 * 3), vc = r / 3, d = r % 3;
    float val;
    if (vc < C_V)             val = ln_v[(size_t)sRow[e] * C_V * 3 + vc * 3 + d];
    else if (vc < C_V + E_VC) val = edge_v[(size_t)(e0 + e) * E_VC * 3 + (vc - C_V) * 3 + d];
    else                      val = ln_v[(size_t)sCol[e] * C_V * 3 + (vc - C_V - E_VC) * 3 + d];
    sMV[e * 108 + r] = val;
  }
  __syncthreads();
  // vh[e,h,d] = sum_v mv[e,v,d] * Wvd[v,h]
  for (int i = t; i < EB0 * H0 * 3; i += 256) {
    int e = i / (H0 * 3), r = i % (H0 * 3), hh = r / 3, d = r % 3;
    float acc = 0.0f;
    for (int v = 0; v < H0; ++v) acc += sMV[e * 108 + v * 3 + d] * sWvd[v * H0 + hh];
    sVH[i] = acc;
  }
  __syncthreads();
  // norms -> sA[:, 288:324]
  for (int i = t; i < EB0 * H0; i += 256) {
    int e = i / H0, hh = i % H0;
    float s2 = 0.0f;
    for (int d = 0; d < 3; ++d) { float x = sVH[e * H0 * 3 + hh * 3 + d]; s2 += x * x; }
    sA[e * SA0 + 288 + hh] = sqrtf(s2 + 1e-8f);
  }
  // frame scalarization -> sA[:, 324:432]
  for (int i = t; i < EB0 * H0 * 3; i += 256) {
    int e = i / (H0 * 3), r = i % (H0 * 3), hh = r / 3, fi = r % 3;
    float acc = 0.0f;
    for (int d = 0; d < 3; ++d)
      acc += sVH[e * H0 * 3 + hh * 3 + d] * frames[(size_t)(e0 + e) * 9 + fi * 3 + d];
    sA[e * SA0 + 288 + H0 + hh * 3 + fi] = acc;
  }
  __syncthreads();
  // WMMA GEMM: (32x432)@(432x128); wave w owns 16 columns, B fragments shared by 2 M-tiles
  const int lane = t & 31, wv = t >> 5;
  const int lo = lane & 15, hi = lane >> 4;
  const int ncol = wv * 16 + lo;
  {
    __builtin_prefetch(Ws + ncol, 0, 3);
    v8f acc[MT0] = {};
    for (int k = 0; k < K0; k += 4) {
      const int ka = k + 2 * hi;
      const float bx = Ws[(size_t)ka * C_S + ncol];
      const float by = Ws[(size_t)(ka + 1) * C_S + ncol];
#pragma unroll
      for (int mt = 0; mt < MT0; ++mt)
        acc[mt] = wmma4(sA[(mt * 16 + lo) * SA0 + ka], sA[(mt * 16 + lo) * SA0 + ka + 1],
                        bx, by, acc[mt]);
    }
    __syncthreads();  // sMV dead; sNS (same LDS) now written
#pragma unroll
    for (int mt = 0; mt < MT0; ++mt) {
#pragma unroll
      for (int i = 0; i < 8; ++i) {
        int m = mt * 16 + 8 * hi + i;
        float x = siluf_(acc[mt][i] + bs[ncol]);
        sNS[m * 129 + ncol] = x;
        rs[(size_t)(e0 + m) * C_S + ncol] = x;
      }
    }
  }
  __syncthreads();
  // gate = sigmoid(ns @ Wg + bg): waves 0..MT0-1 via WMMA, one 16x16 tile each
  if (wv < MT0) {
    v8f acc = {};
    for (int k = 0; k < C_S; k += 4) {
      const int ka = k + 2 * hi;
      acc = wmma4(sNS[(wv * 16 + lo) * 129 + ka], sNS[(wv * 16 + lo) * 129 + ka + 1],
                  Wg[ka * C_V + lo], Wg[(ka + 1) * C_V + lo], acc);
    }
#pragma unroll
    for (int i = 0; i < 8; ++i)
      sGate[(wv * 16 + 8 * hi + i) * 16 + lo] = sigmoidf_(acc[i] + bg[lo]);
  }
  __syncthreads();
  // rv = (vh @ Wvu) * gate
  for (int i = t; i < EB0 * C_V * 3; i += 256) {
    int e = i / (C_V * 3), r = i % (C_V * 3), o = r / 3, d = r % 3;
    float acc = 0.0f;
    for (int hh = 0; hh < H0; ++hh) acc += sVH[e * H0 * 3 + hh * 3 + d] * Wvu[hh * C_V + o];
    rv[(size_t)(e0 + e) * C_V * 3 + r] = acc * sGate[e * 16 + o];
  }
}

// ---------------- Message GCPs 1..3 (fused, residual, 64 edges / block, 4 M-tiles) ----------------
constexpr int H1 = 16;
constexpr int K1 = C_S + 4 * H1;  // 192
constexpr int SA1 = K1 + 4;       // 196
constexpr int EB1 = 64;
constexpr int MT1 = EB1 / 16;

__global__ void __launch_bounds__(256)
k_msg_rest(const float* __restrict__ frames,
           const float* __restrict__ Wvd, const float* __restrict__ Ws,
           const float* __restrict__ bs, const float* __restrict__ Wvu,
           const float* __restrict__ Wg, const float* __restrict__ bg,
           int act, float* __restrict__ rs, float* __restrict__ rv) {
  __shared__ __align__(16) float sA[EB1 * SA1];
  __shared__ float sWvd[H1 * H1];
  __shared__ __align__(16) float sRV[EB1 * H1 * 3];
  __shared__ float sVH[EB1 * H1 * 3];
  __shared__ float sNS[EB1 * 129];
  __shared__ float sGate[EB1 * 16];
  const int t = threadIdx.x;
  const int e0 = blockIdx.x * EB1;
  for (int i = t; i < H1 * H1; i += 256) sWvd[i] = Wvd[i];
  for (int i = t; i < EB1 * (C_S / 4); i += 256) {  // rs rows, float4
    int e = i / (C_S / 4), q = i % (C_S / 4);
    *(float4*)(&sA[e * SA1 + 4 * q]) = *(const float4*)(rs + (size_t)(e0 + e) * C_S + 4 * q);
  }
  for (int i = t; i < EB1 * 12; i += 256)  // rv rows, float4 (48 floats = 12 quads)
    *(float4*)(&sRV[4 * i]) = *(const float4*)(rv + (size_t)e0 * 48 + 4 * i);
  __syncthreads();
  for (int i = t; i < EB1 * H1 * 3; i += 256) {
    int e = i / (H1 * 3), r = i % (H1 * 3), hh = r / 3, d = r % 3;
    float acc = 0.0f;
    for (int v = 0; v < H1; ++v) acc += sRV[e * 48 + v * 3 + d] * sWvd[v * H1 + hh];
    sVH[i] = acc;
  }
  __syncthreads();
  for (int i = t; i < EB1 * H1; i += 256) {
    int e = i / H1, hh = i % H1;
    float s2 = 0.0f;
    for (int d = 0; d < 3; ++d) { float x = sVH[e * 48 + hh * 3 + d]; s2 += x * x; }
    sA[e * SA1 + C_S + hh] = sqrtf(s2 + 1e-8f);
  }
  for (int i = t; i < EB1 * H1 * 3; i += 256) {
    int e = i / (H1 * 3), r = i % (H1 * 3), hh = r / 3, fi = r % 3;
    float acc = 0.0f;
    for (int d = 0; d < 3; ++d)
      acc += sVH[e * 48 + hh * 3 + d] * frames[(size_t)(e0 + e) * 9 + fi * 3 + d];
    sA[e * SA1 + C_S + H1 + hh * 3 + fi] = acc;
  }
  __syncthreads();
  const int lane = t & 31, wv = t >> 5;
  const int lo = lane & 15, hi = lane >> 4;
  const int ncol = wv * 16 + lo;
  {
    v8f acc[MT1] = {};
    for (int k = 0; k < K1; k += 4) {
      const int ka = k + 2 * hi;
      const float bx = Ws[(size_t)ka * C_S + ncol];
      const float by = Ws[(size_t)(ka + 1) * C_S + ncol];
#pragma unroll
      for (int mt = 0; mt < MT1; ++mt)
        acc[mt] = wmma4(sA[(mt * 16 + lo) * SA1 + ka], sA[(mt * 16 + lo) * SA1 + ka + 1],
                        bx, by, acc[mt]);
    }
#pragma unroll
    for (int mt = 0; mt < MT1; ++mt) {
#pragma unroll
      for (int i = 0; i < 8; ++i) {
        int m = mt * 16 + 8 * hi + i;
        float x = acc[mt][i] + bs[ncol];
        if (act) x = siluf_(x);
        sNS[m * 129 + ncol] = x;
        rs[(size_t)(e0 + m) * C_S + ncol] = sA[m * SA1 + ncol] + x;  // residual
      }
    }
  }
  __syncthreads();
  if (wv < MT1) {
    v8f acc = {};
    for (int k = 0; k < C_S; k += 4) {
      const int ka = k + 2 * hi;
      acc = wmma4(sNS[(wv * 16 + lo) * 129 + ka], sNS[(wv * 16 + lo) * 129 + ka + 1],
                  Wg[ka * C_V + lo], Wg[(ka + 1) * C_V + lo], acc);
    }
#pragma unroll
    for (int i = 0; i < 8; ++i)
      sGate[(wv * 16 + 8 * hi + i) * 16 + lo] = sigmoidf_(acc[i] + bg[lo]);
  }
  __syncthreads();
  for (int i = t; i < EB1 * C_V * 3; i += 256) {
    int e = i / (C_V * 3), r = i % (C_V * 3), o = r / 3, d = r % 3;
    float acc = 0.0f;
    for (int hh = 0; hh < H1; ++hh) acc += sVH[e * 48 + hh * 3 + d] * Wvu[hh * C_V + o];
    rv[(size_t)(e0 + e) * C_V * 3 + r] = sRV[e * 48 + r] + acc * sGate[e * 16 + o];
  }
}

// ---------------- edge -> node scatter (sum; divide later) ----------------
__global__ void k_agg(const int* __restrict__ ei, const float* __restrict__ rs,
                      const float* __restrict__ rv, float* __restrict__ agg_s,
                      float* __restrict__ agg_v) {
  long i = (long)blockIdx.x * blockDim.x + threadIdx.x;
  if (i >= (long)N_EDGES * 176) return;
  int e = (int)(i / 176), r = (int)(i % 176);
  int c = ei[N_EDGES + e];
  if (r < C_S) atomicAdd(&agg_s[(size_t)c * C_S + r], rs[(size_t)e * C_S + r]);
  else atomicAdd(&agg_v[(size_t)c * 48 + (r - C_S)], rv[(size_t)e * 48 + (r - C_S)]);
}

__global__ void k_resid(const float* __restrict__ ln_s, const float* __restrict__ ln_v,
                        const float* __restrict__ agg_s, const float* __restrict__ agg_v,
                        const float* __restrict__ cnt_col, float* __restrict__ s1,
                        float* __restrict__ v1) {
  long i = (long)blockIdx.x * blockDim.x + threadIdx.x;
  if (i >= (long)N_NODES * 176) return;
  int n = (int)(i / 176), r = (int)(i % 176);
  float c = fmaxf(cnt_col[n], 1.0f);
  if (r < C_S) s1[(size_t)n * C_S + r] = ln_s[(size_t)n * C_S + r] + agg_s[(size_t)n * C_S + r] / c;
  else v1[(size_t)n * 48 + (r - C_S)] = ln_v[(size_t)n * 48 + (r - C_S)] + agg_v[(size_t)n * 48 + (r - C_S)] / c;
}

// ---------------- FF helpers ----------------
__global__ void k_vh(const float* __restrict__ v_in, int vin, const float* __restrict__ Wvd,
                     int h, float* __restrict__ vh) {
  long i = (long)blockIdx.x * blockDim.x + threadIdx.x;
  if (i >= (long)N_NODES * h * 3) return;
  int node = (int)(i / (h * 3)), r = (int)(i % (h * 3)), hh = r / 3, d = r % 3;
  float acc = 0.0f;
  for (int v = 0; v < vin; ++v) acc += v_in[(size_t)node * vin * 3 + v * 3 + d] * Wvd[v * h + hh];
  vh[(size_t)node * h * 3 + r] = acc;
}

__global__ void k_scal(const int* __restrict__ row, const float* __restrict__ vh, int h,
                       const float* __restrict__ frames, float* __restrict__ acc) {
  long i = (long)blockIdx.x * blockDim.x + threadIdx.x;
  if (i >= (long)N_EDGES * h * 3) return;
  int e = (int)(i / (h * 3)), r = (int)(i % (h * 3)), hh = r / 3, fi = r % 3;
  int nd = row[e];
  float p = 0.0f;
  for (int d = 0; d < 3; ++d)
    p += vh[(size_t)nd * h * 3 + hh * 3 + d] * frames[(size_t)e * 9 + fi * 3 + d];
  atomicAdd(&acc[(size_t)nd * h * 3 + r], p);
}

// node GEMM: s_cat = [s | norms(vh) | scal/cnt] (32 nodes / block, 2 M-tiles / wave)
__global__ void __launch_bounds__(256)
k_ff_gemm(const float* __restrict__ s_in, int sdim, const float* __restrict__ vh, int h,
          const float* __restrict__ scal_acc, const float* __restrict__ cnt_row,
          const float* __restrict__ Ws, const float* __restrict__ bs, int ncols, int act,
          float* __restrict__ out) {
  __shared__ __align__(16) float sA[32 * 644];  // max: K=640 -> stride 644
  const int t = threadIdx.x;
  const int n0node = blockIdx.x * 32;
  const int K = sdim + 4 * h;
  const int SA = K + 4;
  const int h3 = h * 3;
  for (int i = t; i < 32 * (sdim / 4); i += 256) {  // scalar part, float4
    int r = i / (sdim / 4), q = i % (sdim / 4);
    *(float4*)(&sA[r * SA + 4 * q]) = *(const float4*)(s_in + (size_t)(n0node + r) * sdim + 4 * q);
  }
  for (int i = t; i < 32 * 4 * h; i += 256) {  // norms + scal part
    int r = i / (4 * h), c2 = i % (4 * h);
    int node = n0node + r;
    float val;
    if (c2 < h) {
      float s2 = 0.0f;
      for (int d = 0; d < 3; ++d) { float x = vh[(size_t)node * h3 + c2 * 3 + d]; s2 += x * x; }
      val = sqrtf(s2 + 1e-8f);
    } else {
      float cr = fmaxf(cnt_row[node], 1.0f);
      val = scal_acc[(size_t)node * h3 + (c2 - h)] / cr;
    }
    sA[r * SA + sdim + c2] = val;
  }
  __syncthreads();
  const int lane = t & 31, wv = t >> 5;
  const int lo = lane & 15, hi = lane >> 4;
  const int ncol = (blockIdx.y * 8 + wv) * 16 + lo;
  v8f acc[2] = {};
  for (int k = 0; k < K; k += 4) {
    const int ka = k + 2 * hi;
    const float bx = Ws[(size_t)ka * ncols + ncol];
    const float by = Ws[(size_t)(ka + 1) * ncols + ncol];
#pragma unroll
    for (int mt = 0; mt < 2; ++mt)
      acc[mt] = wmma4(sA[(mt * 16 + lo) * SA + ka], sA[(mt * 16 + lo) * SA + ka + 1],
                      bx, by, acc[mt]);
  }
#pragma unroll
  for (int mt = 0; mt < 2; ++mt) {
#pragma unroll
    for (int i = 0; i < 8; ++i) {
      int m = mt * 16 + 8 * hi + i;
      float x = acc[mt][i] + bs[ncol];
      if (act) x = siluf_(x);
      out[(size_t)(n0node + m) * ncols + ncol] = x;
    }
  }
}

__global__ void k_gate_vout(const float* __restrict__ s_act, int sdim,
                            const float* __restrict__ Wg, const float* __restrict__ bg,
                            const float* __restrict__ vh, int h, const float* __restrict__ Wvu,
                            int vout, float* __restrict__ vdst) {
  long i = (long)blockIdx.x * blockDim.x + threadIdx.x;
  if (i >= (long)N_NODES * vout) return;
  int node = (int)(i / vout), o = (int)(i % vout);
  float acc = bg[o];
  const float* sp = s_act + (size_t)node * sdim;
  for (int k = 0; k < sdim; ++k) acc += sp[k] * Wg[k * vout + o];
  float gate = sigmoidf_(acc);
  const float* vp = vh + (size_t)node * h * 3;
  for (int d = 0; d < 3; ++d) {
    float a = 0.0f;
    for (int hh = 0; hh < h; ++hh) a += vp[hh * 3 + d] * Wvu[hh * vout + o];
    vdst[(size_t)node * vout * 3 + o * 3 + d] = a * gate;
  }
}

__global__ void k_final(const float* __restrict__ s1, const float* __restrict__ v1,
                        const float* __restrict__ hs, const float* __restrict__ hv,
                        float* __restrict__ out) {
  long i = (long)blockIdx.x * blockDim.x + threadIdx.x;
  if (i >= (long)N_NODES * 176) return;
  int n = (int)(i / 176), r = (int)(i % 176);
  if (r < C_S) out[(size_t)n * C_S + r] = s1[(size_t)n * C_S + r] + hs[(size_t)n * C_S + r];
  else out[(size_t)N_NODES * C_S + (size_t)n * 48 + (r - C_S)] =
      v1[(size_t)n * 48 + (r - C_S)] + hv[(size_t)n * 48 + (r - C_S)];
}

extern "C" void kernel_launch(void* const* d_in, const int* in_sizes, int n_in, void* d_out,
                              int out_size, void* d_ws, size_t ws_size, hipStream_t stream) {
  (void)in_sizes; (void)n_in; (void)out_size; (void)ws_size;
  const float* node_s = (const float*)d_in[0];
  const float* node_v = (const float*)d_in[1];
  const float* edge_s = (const float*)d_in[2];
  const float* edge_v = (const float*)d_in[3];
  const float* frames = (const float*)d_in[4];
  struct GCPP { const float *Wvd, *Ws, *bs, *Wvu, *Wg, *bg; };
  int pi = 5;
  GCPP msg[4], ff[2];
  for (int i = 0; i < 4; ++i) {
    msg[i].Wvd = (const float*)d_in[pi++]; msg[i].Ws = (const float*)d_in[pi++];
    msg[i].bs = (const float*)d_in[pi++];  msg[i].Wvu = (const float*)d_in[pi++];
    msg[i].Wg = (const float*)d_in[pi++];  msg[i].bg = (const float*)d_in[pi++];
  }
  for (int i = 0; i < 2; ++i) {
    ff[i].Wvd = (const float*)d_in[pi++]; ff[i].Ws = (const float*)d_in[pi++];
    ff[i].bs = (const float*)d_in[pi++];  ff[i].Wvu = (const float*)d_in[pi++];
    ff[i].Wg = (const float*)d_in[pi++];  ff[i].bg = (const float*)d_in[pi++];
  }
  const float* ln0g = (const float*)d_in[pi++];
  const float* ln0b = (const float*)d_in[pi++];
  const float* ln1g = (const float*)d_in[pi++];
  const float* ln1b = (const float*)d_in[pi++];
  const int* ei = (const int*)d_in[pi++];

  float* W = (float*)d_ws;
  size_t off = 0;
  auto alloc = [&](size_t n) { float* p = W + off; off += n; return p; };
  float* ln_s = alloc((size_t)N_NODES * C_S);
  float* ln_v = alloc((size_t)N_NODES * 48);
  float* rs = alloc((size_t)N_EDGES * C_S);
  float* rv = alloc((size_t)N_EDGES * 48);
  float* agg_s = alloc((size_t)N_NODES * C_S);
  float* agg_v = alloc((size_t)N_NODES * 48);
  float* cnt_row = alloc(N_NODES);
  float* cnt_col = alloc(N_NODES);
  float* s1 = alloc((size_t)N_NODES * C_S);
  float* v1 = alloc((size_t)N_NODES * 48);
  float* h_s = alloc((size_t)N_NODES * C_S);
  float* h_v = alloc((size_t)N_NODES * 48);
  float* vh_ff = alloc((size_t)N_NODES * 96);
  float* scal_acc = alloc((size_t)N_NODES * 96);
  float* hs1 = alloc((size_t)N_NODES * 512);
  float* hv1 = alloc((size_t)N_NODES * 96);
  float* hs2 = alloc((size_t)N_NODES * C_S);
  float* hv2 = alloc((size_t)N_NODES * 48);

  auto cdiv = [](long a, long b) { return (unsigned)((a + b - 1) / b); };
  auto zero = [&](float* p, long n) { k_zero<<<cdiv(n, 256), 256, 0, stream>>>(p, n); };

  // ln0
  k_layernorm<<<N_NODES, C_S, 0, stream>>>(node_s, node_v, ln0g, ln0b, ln_s, ln_v);
  // counts
  zero(cnt_row, N_NODES); zero(cnt_col, N_NODES);
  k_counts<<<cdiv(N_EDGES, 256), 256, 0, stream>>>(ei, cnt_row, cnt_col);
  // messages
  k_msg0<<<N_EDGES / EB0, 256, 0, stream>>>(ln_s, ln_v, edge_s, edge_v, frames, ei,
                                            msg[0].Wvd, msg[0].Ws, msg[0].bs, msg[0].Wvu,
                                            msg[0].Wg, msg[0].bg, rs, rv);
  for (int i = 1; i < 4; ++i) {
    int act = (i < 3) ? 1 : 0;  // last message GCP has no scalar nonlinearity
    k_msg_rest<<<N_EDGES / EB1, 256, 0, stream>>>(frames, msg[i].Wvd, msg[i].Ws, msg[i].bs,
                                                  msg[i].Wvu, msg[i].Wg, msg[i].bg, act, rs, rv);
  }
  // aggregate (scatter-mean over col) + residual
  zero(agg_s, (long)N_NODES * C_S); zero(agg_v, (long)N_NODES * 48);
  k_agg<<<cdiv((long)N_EDGES * 176, 256), 256, 0, stream>>>(ei, rs, rv, agg_s, agg_v);
  k_resid<<<cdiv((long)N_NODES * 176, 256), 256, 0, stream>>>(ln_s, ln_v, agg_s, agg_v, cnt_col,
                                                              s1, v1);
  // ln1
  k_layernorm<<<N_NODES, C_S, 0, stream>>>(s1, v1, ln1g, ln1b, h_s, h_v);
  // FF GCP 0: s_in=128, h=32, s_out=512, v_out=32, act
  k_vh<<<cdiv((long)N_NODES * 96, 256), 256, 0, stream>>>(h_v, 16, ff[0].Wvd, 32, vh_ff);
  zero(scal_acc, (long)N_NODES * 96);
  k_scal<<<cdiv((long)N_EDGES * 96, 256), 256, 0, stream>>>(ei, vh_ff, 32, frames, scal_acc);
  k_ff_gemm<<<dim3(N_NODES / 32, 4), 256, 0, stream>>>(h_s, C_S, vh_ff, 32, scal_acc, cnt_row,
                                                       ff[0].Ws, ff[0].bs, 512, 1, hs1);
  k_gate_vout<<<cdiv((long)N_NODES * 32, 256), 256, 0, stream>>>(hs1, 512, ff[0].Wg, ff[0].bg,
                                                                 vh_ff, 32, ff[0].Wvu, 32, hv1);
  // FF GCP 1: s_in=512, h=32, s_out=128, v_out=16, no act
  k_vh<<<cdiv((long)N_NODES * 96, 256), 256, 0, stream>>>(hv1, 32, ff[1].Wvd, 32, vh_ff);
  zero(scal_acc, (long)N_NODES * 96);
  k_scal<<<cdiv((long)N_EDGES * 96, 256), 256, 0, stream>>>(ei, vh_ff, 32, frames, scal_acc);
  k_ff_gemm<<<dim3(N_NODES / 32, 1), 256, 0, stream>>>(hs1, 512, vh_ff, 32, scal_acc, cnt_row,
                                                       ff[1].Ws, ff[1].bs, C_S, 0, hs2);
  k_gate_vout<<<cdiv((long)N_NODES * 16, 256), 256, 0, stream>>>(hs2, C_S, ff[1].Wg, ff[1].bg,
                                                                 vh_ff, 32, ff[1].Wvu, 16, hv2);
  // final residual -> d_out  ([N,128] then [N,16,3])
  k_final<<<cdiv((long)N_NODES * 176, 256), 256, 0, stream>>>(s1, v1, hs2, hv2, (float*)d_out);
}